// Cutout_39822936768614
// MI455X (gfx1250) — compile-verified
//
#include <hip/hip_runtime.h>

namespace {
constexpr int Bn = 64, Cn = 3, Hn = 512, Wn = 512, Kn = 5;
constexpr int HW = Hn * Wn;            // 262144 floats per plane
constexpr int V4_PER_PLANE = HW / 4;   // 65536 float4 per plane
constexpr int THREADS = 256;
}

// Native clang vector type: accepted by the nontemporal builtins and lowers
// to 128-bit global loads/stores.
typedef __attribute__((ext_vector_type(4))) float v4f;

// One thread owns one float4 position (b, y, x0..x0+3) and applies the mask to
// all 3 channels: mask computed once, 96 bytes moved. Pure streaming kernel;
// target is the 23.3 TB/s HBM roofline (~17 us for 403 MB).
__global__ __launch_bounds__(THREADS, 8) void cutout_kernel(
    const float* __restrict__ img,
    const int*   __restrict__ num_holes,
    const int*   __restrict__ ys,
    const int*   __restrict__ xs,
    const int*   __restrict__ hs,
    const int*   __restrict__ ws,
    float*       __restrict__ out)
{
    const int b   = blockIdx.y;                               // uniform -> SMEM param loads
    const int off = blockIdx.x * THREADS + threadIdx.x;       // float4 index within a plane
    const int y   = off >> 7;                                 // W/4 = 128 float4 per row
    const int x0  = (off & 127) << 2;

    const int nh = num_holes[b];

    bool kill0 = false, kill1 = false, kill2 = false, kill3 = false;
#pragma unroll
    for (int k = 0; k < Kn; ++k) {
        const int idx = b * Kn + k;
        const int yc = ys[idx];
        const int xc = xs[idx];
        const int hh = hs[idx] >> 1;   // hs >= 1, floor-div by 2
        const int hw = ws[idx] >> 1;

        // clamp to [0, H] / [0, W], matching jnp.clip semantics
        const int y1 = min(max(yc - hh, 0), Hn);
        const int y2 = min(max(yc + hh, 0), Hn);
        const int x1 = min(max(xc - hw, 0), Wn);
        const int x2 = min(max(xc + hw, 0), Wn);

        const bool iny = (k < nh) & (y >= y1) & (y < y2);
        const unsigned span = (unsigned)(x2 - x1);            // x2 >= x1 after clamp
        kill0 |= iny & ((unsigned)(x0 + 0 - x1) < span);
        kill1 |= iny & ((unsigned)(x0 + 1 - x1) < span);
        kill2 |= iny & ((unsigned)(x0 + 2 - x1) < span);
        kill3 |= iny & ((unsigned)(x0 + 3 - x1) < span);
    }

    const size_t base = (size_t)b * (size_t)(Cn * HW) + (size_t)off * 4;
#pragma unroll
    for (int c = 0; c < Cn; ++c) {
        const v4f v = __builtin_nontemporal_load(
            reinterpret_cast<const v4f*>(img + base + (size_t)c * HW));
        v4f r;
        r.x = kill0 ? 0.0f : v.x;
        r.y = kill1 ? 0.0f : v.y;
        r.z = kill2 ? 0.0f : v.z;
        r.w = kill3 ? 0.0f : v.w;
        __builtin_nontemporal_store(r,
            reinterpret_cast<v4f*>(out + base + (size_t)c * HW));
    }
}

extern "C" void kernel_launch(void* const* d_in, const int* in_sizes, int n_in,
                              void* d_out, int out_size, void* d_ws, size_t ws_size,
                              hipStream_t stream) {
    (void)in_sizes; (void)n_in; (void)out_size; (void)d_ws; (void)ws_size;
    const float* img       = (const float*)d_in[0];
    const int*   num_holes = (const int*)d_in[1];
    const int*   ys        = (const int*)d_in[2];
    const int*   xs        = (const int*)d_in[3];
    const int*   hs        = (const int*)d_in[4];
    const int*   ws        = (const int*)d_in[5];
    float*       out       = (float*)d_out;

    dim3 grid(V4_PER_PLANE / THREADS, Bn);   // (256, 64)
    cutout_kernel<<<grid, dim3(THREADS), 0, stream>>>(
        img, num_holes, ys, xs, hs, ws, out);
}